// SinkhornOT_57964878627353
// MI455X (gfx1250) — compile-verified
//
#include <hip/hip_runtime.h>
#include <hip/hip_bf16.h>

typedef __attribute__((ext_vector_type(2))) float v2f;
typedef __attribute__((ext_vector_type(8))) float v8f;

#define B_  32
#define N_  16384
#define K_  64
#define D_  256
#define NSPLIT 64           // column-reduction segments over N
#define INV_EPS 20.0f       // 1/0.05
#define LOG_NU (-4.1588830833596715f)  // -log(64)

// ---------------------------------------------------------------------------
// Kernel 1: fused batched GEMM (cross = P·S^T) via V_WMMA_F32_16X16X4_F32,
// with in-register row/col squared norms, writing log_K = max(-C/eps, -100)
// into d_out (used as the 134MB log_K scratch; it stays L2-resident).
// One wave computes a 16x64 strip (4 accumulators of 16x16).
// ---------------------------------------------------------------------------
__global__ void __launch_bounds__(256) gemm_logk_kernel(
    const float* __restrict__ P, const float* __restrict__ S,
    float* __restrict__ logK) {
  const int wave = (blockIdx.x * 256 + threadIdx.x) >> 5;  // 32768 waves total
  const int lane = threadIdx.x & 31;
  const int b  = wave >> 10;        // 1024 row-tiles per batch
  const int n0 = (wave & 1023) << 4;

  const float* Pb = P + ((size_t)b * N_) * D_;
  const float* Sb = S + ((size_t)b * K_) * D_;

  const int mrow = lane & 15;
  const int koff = (lane < 16) ? 0 : 2;   // fp32 WMMA A/B: half-waves split K

  // A tile rows: pixel row n0+mrow; B tile cols: slot t*16+mrow
  const float* arow = Pb + (size_t)(n0 + mrow) * D_ + koff;
  const float* bcol = Sb + (size_t)mrow * D_ + koff;

  v8f acc0 = {}, acc1 = {}, acc2 = {}, acc3 = {};
  float pa = 0.f, q0 = 0.f, q1 = 0.f, q2 = 0.f, q3 = 0.f;

  for (int d0 = 0; d0 < D_; d0 += 4) {
    v2f a  = *(const v2f*)(arow + d0);
    v2f b0 = *(const v2f*)(bcol + d0);
    v2f b1 = *(const v2f*)(bcol + 16 * D_ + d0);
    v2f b2 = *(const v2f*)(bcol + 32 * D_ + d0);
    v2f b3 = *(const v2f*)(bcol + 48 * D_ + d0);
    acc0 = __builtin_amdgcn_wmma_f32_16x16x4_f32(false, a, false, b0, (short)0, acc0, false, false);
    acc1 = __builtin_amdgcn_wmma_f32_16x16x4_f32(false, a, false, b1, (short)0, acc1, false, false);
    acc2 = __builtin_amdgcn_wmma_f32_16x16x4_f32(false, a, false, b2, (short)0, acc2, false, false);
    acc3 = __builtin_amdgcn_wmma_f32_16x16x4_f32(false, a, false, b3, (short)0, acc3, false, false);
    pa += a.x * a.x + a.y * a.y;          // partial ||p||^2 (half of K-dim)
    q0 += b0.x * b0.x + b0.y * b0.y;      // partial ||s||^2 per column
    q1 += b1.x * b1.x + b1.y * b1.y;
    q2 += b2.x * b2.x + b2.y * b2.y;
    q3 += b3.x * b3.x + b3.y * b3.y;
  }
  // combine the two half-wave K partitions
  pa += __shfl_xor(pa, 16, 32);
  q0 += __shfl_xor(q0, 16, 32);
  q1 += __shfl_xor(q1, 16, 32);
  q2 += __shfl_xor(q2, 16, 32);
  q3 += __shfl_xor(q3, 16, 32);

  float* out = logK + ((size_t)b * N_ + n0) * K_;
  const int hi = (lane >= 16) ? 8 : 0;
#pragma unroll
  for (int j = 0; j < 8; ++j) {
    const int ml = j + hi;                 // local row of this C/D element
    const float psq = __shfl(pa, ml, 32);  // ||p||^2 lives in lane ml
    float c0 = psq + q0 - 2.0f * acc0[j];
    float c1 = psq + q1 - 2.0f * acc1[j];
    float c2 = psq + q2 - 2.0f * acc2[j];
    float c3 = psq + q3 - 2.0f * acc3[j];
    float* r = out + (size_t)ml * K_ + mrow;
    r[0]  = fmaxf(-c0 * INV_EPS, -100.0f);
    r[16] = fmaxf(-c1 * INV_EPS, -100.0f);
    r[32] = fmaxf(-c2 * INV_EPS, -100.0f);
    r[48] = fmaxf(-c3 * INV_EPS, -100.0f);
  }
}

// ---------------------------------------------------------------------------
// Kernel 0: zero log_u / log_v (d_ws is poisoned; iterations start from 0)
// ---------------------------------------------------------------------------
__global__ void init_uv_kernel(float* __restrict__ log_u, float* __restrict__ log_v) {
  int i = blockIdx.x * 256 + threadIdx.x;
  if (i < B_ * N_) log_u[i] = 0.0f;
  if (i < B_ * K_) log_v[i] = 0.0f;
}

// ---------------------------------------------------------------------------
// Row update: log_u[b,n] = log_mu[b,n] - LSE_k(log_K[b,n,k] + log_v[b,k])
// One thread per row: 64 contiguous floats via 16x float4 (L2-resident).
// ---------------------------------------------------------------------------
__global__ void __launch_bounds__(256) row_update_kernel(
    const float* __restrict__ logK, const float* __restrict__ sw,
    const float* __restrict__ log_v, float* __restrict__ log_u) {
  __shared__ float lv[K_];
  const int b = blockIdx.y;
  if (threadIdx.x < K_) lv[threadIdx.x] = log_v[b * K_ + threadIdx.x];
  __syncthreads();
  const int n = blockIdx.x * 256 + threadIdx.x;
  const float4* row = (const float4*)(logK + ((size_t)b * N_ + n) * K_);
  float4 x[16];
#pragma unroll
  for (int i = 0; i < 16; ++i) x[i] = row[i];
  float m = -1e30f;
#pragma unroll
  for (int i = 0; i < 16; ++i) {
    m = fmaxf(m, fmaxf(fmaxf(x[i].x + lv[4 * i + 0], x[i].y + lv[4 * i + 1]),
                       fmaxf(x[i].z + lv[4 * i + 2], x[i].w + lv[4 * i + 3])));
  }
  float s = 0.0f;
#pragma unroll
  for (int i = 0; i < 16; ++i) {
    s += __expf(x[i].x + lv[4 * i + 0] - m) + __expf(x[i].y + lv[4 * i + 1] - m) +
         __expf(x[i].z + lv[4 * i + 2] - m) + __expf(x[i].w + lv[4 * i + 3] - m);
  }
  const float lse = m + __logf(s);
  const float lmu = __logf(fmaxf(sw[(size_t)b * N_ + n], 1e-8f));
  log_u[(size_t)b * N_ + n] = lmu - lse;
}

// ---------------------------------------------------------------------------
// Column LSE, stage 1: online logsumexp over a 256-row segment per block.
// Block: 256 threads = 4 row-groups x 64 columns (coalesced 1KB/step).
// ---------------------------------------------------------------------------
__global__ void __launch_bounds__(256) col_partial_kernel(
    const float* __restrict__ logK, const float* __restrict__ log_u,
    float* __restrict__ pm, float* __restrict__ ps) {
  const int RPB = N_ / NSPLIT;  // 256 rows per block
  const int b = blockIdx.y, sb = blockIdx.x;
  const int k = threadIdx.x & 63;
  const int g = threadIdx.x >> 6;  // 0..3
  const int n0 = sb * RPB;
  float m = -1e30f, s = 0.0f;
  for (int i = 0; i < RPB / 4; ++i) {
    const int n = n0 + i * 4 + g;
    const float x = logK[((size_t)b * N_ + n) * K_ + k] + log_u[(size_t)b * N_ + n];
    const float nm = fmaxf(m, x);
    s = s * __expf(m - nm) + __expf(x - nm);
    m = nm;
  }
  __shared__ float sm[4][K_], ss[4][K_];
  sm[g][k] = m;
  ss[g][k] = s;
  __syncthreads();
  if (threadIdx.x < K_) {
    float gm = sm[0][k], gs = ss[0][k];
#pragma unroll
    for (int g2 = 1; g2 < 4; ++g2) {
      const float m2 = sm[g2][k], s2 = ss[g2][k];
      const float nm = fmaxf(gm, m2);
      gs = gs * __expf(gm - nm) + s2 * __expf(m2 - nm);
      gm = nm;
    }
    const size_t idx = ((size_t)b * NSPLIT + sb) * K_ + k;
    pm[idx] = gm;
    ps[idx] = gs;
  }
}

// Column LSE, stage 2: merge 64 partials -> log_v[b,k]
__global__ void col_final_kernel(const float* __restrict__ pm,
                                 const float* __restrict__ ps,
                                 float* __restrict__ log_v) {
  const int b = blockIdx.x, k = threadIdx.x;  // 64 threads
  float m = -1e30f, s = 0.0f;
  for (int i = 0; i < NSPLIT; ++i) {
    const size_t idx = ((size_t)b * NSPLIT + i) * K_ + k;
    const float m2 = pm[idx], s2 = ps[idx];
    const float nm = fmaxf(m, m2);
    s = s * __expf(m - nm) + s2 * __expf(m2 - nm);
    m = nm;
  }
  log_v[b * K_ + k] = LOG_NU - (m + __logf(s));
}

// ---------------------------------------------------------------------------
// Finalize in place: d_out = exp(log_u + log_K + log_v)
// ---------------------------------------------------------------------------
__global__ void __launch_bounds__(256) finalize_kernel(
    float* __restrict__ logK, const float* __restrict__ log_u,
    const float* __restrict__ log_v) {
  __shared__ float lv[K_];
  const int b = blockIdx.y;
  if (threadIdx.x < K_) lv[threadIdx.x] = log_v[b * K_ + threadIdx.x];
  __syncthreads();
  const int n = blockIdx.x * 256 + threadIdx.x;
  const float lu = log_u[(size_t)b * N_ + n];
  float4* row = (float4*)(logK + ((size_t)b * N_ + n) * K_);
#pragma unroll
  for (int i = 0; i < 16; ++i) {
    float4 x = row[i];
    x.x = __expf(x.x + lu + lv[4 * i + 0]);
    x.y = __expf(x.y + lu + lv[4 * i + 1]);
    x.z = __expf(x.z + lu + lv[4 * i + 2]);
    x.w = __expf(x.w + lu + lv[4 * i + 3]);
    row[i] = x;
  }
}

extern "C" void kernel_launch(void* const* d_in, const int* in_sizes, int n_in,
                              void* d_out, int out_size, void* d_ws, size_t ws_size,
                              hipStream_t stream) {
  const float* P  = (const float*)d_in[0];  // pixel_feats (B,N,D)
  const float* S  = (const float*)d_in[1];  // slot_feats  (B,K,D)
  const float* SW = (const float*)d_in[2];  // source_weights (B,N)
  float* logK = (float*)d_out;              // (B,N,K): scratch then final output

  float* ws    = (float*)d_ws;
  float* log_u = ws;                               // B*N   = 524288
  float* log_v = log_u + (size_t)B_ * N_;          // B*K   = 2048
  float* pm    = log_v + (size_t)B_ * K_;          // B*NSPLIT*K = 131072
  float* ps    = pm + (size_t)B_ * NSPLIT * K_;    // 131072

  init_uv_kernel<<<(B_ * N_ + 255) / 256, 256, 0, stream>>>(log_u, log_v);
  gemm_logk_kernel<<<(B_ * (N_ / 16)) / 8, 256, 0, stream>>>(P, S, logK);

  const dim3 gridRows(N_ / 256, B_);
  for (int it = 0; it < 20; ++it) {
    row_update_kernel<<<gridRows, 256, 0, stream>>>(logK, SW, log_v, log_u);
    col_partial_kernel<<<dim3(NSPLIT, B_), 256, 0, stream>>>(logK, log_u, pm, ps);
    col_final_kernel<<<B_, K_, 0, stream>>>(pm, ps, log_v);
  }
  finalize_kernel<<<gridRows, 256, 0, stream>>>(logK, log_u, log_v);
}